// NewtonImplicitNet_65635690217645
// MI455X (gfx1250) — compile-verified
//
#include <hip/hip_runtime.h>
#include <hip/hip_bf16.h>
#include <math.h>

typedef float v2f __attribute__((ext_vector_type(2)));
typedef float v8f __attribute__((ext_vector_type(8)));

#define UNITS   128
#define DIN     784
#define BATCH   2048
#define DOUT    10
#define ITERS   160      // even; contraction ~0.5-0.8 -> residual << 1e-6
#define ROWS_WG 16       // one WMMA M-strip per workgroup
#define NTHR    256      // 8 waves: one 16-col tile per wave
#define LSTR    132      // padded LDS row stride (floats): conflict-free fragments

// Hardware tanh (V_TANH_F32) when the builtin exists, else libm fallback.
#if defined(__has_builtin)
# if __has_builtin(__builtin_amdgcn_tanhf)
#   define TANH(v) __builtin_amdgcn_tanhf(v)
# elif __has_builtin(__builtin_amdgcn_tanh_f32)
#   define TANH(v) __builtin_amdgcn_tanh_f32(v)
# else
#   define TANH(v) tanhf(v)
# endif
#else
# define TANH(v) tanhf(v)
#endif

__device__ __forceinline__ v8f wmma4(v2f a, v2f b, v8f c) {
  // D = A(16x4 f32) * B(4x16 f32) + C(16x16 f32)
  return __builtin_amdgcn_wmma_f32_16x16x4_f32(
      /*neg_a=*/false, a, /*neg_b=*/false, b,
      /*c_mod=*/(short)0, c, /*reuse_a=*/false, /*reuse_b=*/false);
}

__launch_bounds__(NTHR, 1)
__global__ void newton_net_kernel(
    const float* __restrict__ x,
    const float* __restrict__ W_in,  const float* __restrict__ b_in,
    const float* __restrict__ W1,    const float* __restrict__ W2,
    const float* __restrict__ W3,    const float* __restrict__ W4,
    const float* __restrict__ W_out, const float* __restrict__ b_out,
    float* __restrict__ out)
{
  __shared__ float lds_W[UNITS * LSTR];       // 67.6 KB  current layer W, row-major
  __shared__ float lds_z0[ROWS_WG * LSTR];    //  8.4 KB  Picard iterate (ping)
  __shared__ float lds_z1[ROWS_WG * LSTR];    //  8.4 KB  Picard iterate (pong)

  const int tid   = threadIdx.x;
  const int lane  = tid & 31;
  const int wave  = tid >> 5;          // wave w owns column tile n0 = 16*w
  const int lmod  = lane & 15;         // M (A) / N (B,C) within tile
  const int lhalf = lane >> 4;         // k-pair select (A,B) / +8 row select (C)
  const int grow0 = blockIdx.x * ROWS_WG;     // global batch row base of strip
  const int n     = wave * 16 + lmod;         // output column of this lane

  // ---- Stage 1: xf = (x @ W_in^T + b_in) tile, kept in registers (C layout)
  float xf[8];
  {
    const float* xrow = x    + (size_t)(grow0 + lmod) * DIN + 2 * lhalf;  // A frag
    const float* wrow = W_in + (size_t)n * DIN + 2 * lhalf;               // B frag
    v8f acc = {};
    #pragma unroll 4
    for (int k = 0; k < DIN; k += 4) {
      v2f a = *(const v2f*)(xrow + k);
      v2f b = *(const v2f*)(wrow + k);
      acc = wmma4(a, b, acc);
    }
    const float bias = b_in[n];
    #pragma unroll
    for (int v = 0; v < 8; ++v) xf[v] = acc[v] + bias;
  }

  const float* Ws[4] = {W1, W2, W3, W4};
  float znew[8];

  // ---- Stage 2: four implicit blocks via Picard fixed-point iteration ----
  #pragma unroll 1
  for (int layer = 0; layer < 4; ++layer) {
    __syncthreads();                                   // lds_W safe to overwrite
    {
      const float* Wl = Ws[layer];
      for (int i = tid; i < UNITS * (UNITS / 4); i += NTHR) {
        const int r = i >> 5, q = i & 31;              // row, float4 chunk
        ((float4*)&lds_W[r * LSTR])[q] = ((const float4*)(Wl + (size_t)r * UNITS))[q];
      }
      if (layer < 3) {                                 // warm L2/WGP$ for next W
        const char* nw = (const char*)Ws[layer + 1];
        __builtin_prefetch(nw + tid * 256,       0, 1);
        __builtin_prefetch(nw + tid * 256 + 128, 0, 1);
      }
    }
    __syncthreads();

    // Hoist the 32 invariant B fragments of W^T into registers.
    v2f bf[32];
    {
      const float* wb = &lds_W[n * LSTR + 2 * lhalf];  // B[k][n] = W[n][k]
      #pragma unroll
      for (int k = 0; k < 32; ++k) bf[k] = *(const v2f*)(wb + 4 * k);
    }

    // z0 <- tanh(x): each wave writes its own C-layout tile (covers full strip).
    #pragma unroll
    for (int v = 0; v < 8; ++v)
      lds_z0[(v + 8 * lhalf) * LSTR + n] = TANH(xf[v]);
    __syncthreads();

    #pragma unroll 1
    for (int it = 0; it < ITERS; ++it) {
      const float* zr = (it & 1) ? lds_z1 : lds_z0;    // read buffer
      float*       zw = (it & 1) ? lds_z0 : lds_z1;    // write buffer
      const float* za = zr + lmod * LSTR + 2 * lhalf;  // A frag base (rows of strip)
      v8f acc = {};
      #pragma unroll
      for (int k = 0; k < 32; ++k) {
        v2f a = *(const v2f*)(za + 4 * k);
        acc = wmma4(a, bf[k], acc);
      }
      #pragma unroll
      for (int v = 0; v < 8; ++v) znew[v] = TANH(acc[v] + xf[v]);
      #pragma unroll
      for (int v = 0; v < 8; ++v) zw[(v + 8 * lhalf) * LSTR + n] = znew[v];
      __syncthreads();                                 // ping-pong: 1 barrier/iter
    }

    // Block output tanh(z* W^T + x) == last iterate; it is this wave's znew tile.
    #pragma unroll
    for (int v = 0; v < 8; ++v) xf[v] = znew[v];
  }

  // ---- Stage 3: out = softmax(z @ W_out^T + b_out), one row per lane --------
  // Final z strip lives in lds_z0 (ITERS even: last write hit ping buffer).
  if (tid < ROWS_WG) {
    const int grow = grow0 + tid;
    float logits[DOUT];
    #pragma unroll
    for (int o = 0; o < DOUT; ++o) {
      float acc = b_out[o];
      #pragma unroll 4
      for (int u = 0; u < UNITS; ++u)
        acc += lds_z0[tid * LSTR + u] * W_out[(size_t)o * UNITS + u];
      logits[o] = acc;
    }
    float m = logits[0];
    #pragma unroll
    for (int o = 1; o < DOUT; ++o) m = fmaxf(m, logits[o]);
    float s = 0.0f, e[DOUT];
    #pragma unroll
    for (int o = 0; o < DOUT; ++o) { e[o] = expf(logits[o] - m); s += e[o]; }
    const float inv = 1.0f / s;
    #pragma unroll
    for (int o = 0; o < DOUT; ++o)
      out[(size_t)grow * DOUT + o] = e[o] * inv;
  }
}

extern "C" void kernel_launch(void* const* d_in, const int* in_sizes, int n_in,
                              void* d_out, int out_size, void* d_ws, size_t ws_size,
                              hipStream_t stream) {
  const float* x     = (const float*)d_in[0];
  const float* W_in  = (const float*)d_in[1];
  const float* b_in  = (const float*)d_in[2];
  const float* W1    = (const float*)d_in[3];
  const float* W2    = (const float*)d_in[4];
  const float* W3    = (const float*)d_in[5];
  const float* W4    = (const float*)d_in[6];
  const float* W_out = (const float*)d_in[7];
  const float* b_out = (const float*)d_in[8];

  dim3 grid(BATCH / ROWS_WG);   // 128 workgroups (one 16-row strip each)
  dim3 block(NTHR);             // 8 waves: one 16-col tile per wave
  hipLaunchKernelGGL(newton_net_kernel, grid, block, 0, stream,
                     x, W_in, b_in, W1, W2, W3, W4, W_out, b_out,
                     (float*)d_out);
}